// MultiSynonymsAttention_82849919140625
// MI455X (gfx1250) — compile-verified
//
#include <hip/hip_runtime.h>

// ---------------------------------------------------------------------------
// MultiSynonymsAttention for MI455X (gfx1250), f32 WMMA + TDM path.
// d_model=768, heads z=12, d_k h=64, classes c=50, synonyms s=4, b=4, t=1024
// ---------------------------------------------------------------------------

typedef __attribute__((ext_vector_type(2))) float        v2f;
typedef __attribute__((ext_vector_type(8))) float        v8f;
typedef __attribute__((ext_vector_type(4))) unsigned int v4u;
typedef __attribute__((ext_vector_type(8))) int          v8i;
typedef __attribute__((ext_vector_type(4))) int          v4i;

#define D_MODEL 768
#define NZ 12
#define NH 64
#define NC 50
#define NS 4
#define NBATCH 4
#define NT 1024
#define CS (NC * NS)   /* 200 rows of (c,s) */
#define SLD 1028       /* padded LDS stride for score rows (bank-conflict free) */

static __device__ __forceinline__ v8f wmma_f32(v2f a, v2f b, v8f c) {
  // V_WMMA_F32_16X16X4_F32 : D(16x16,f32) = A(16x4,f32) * B(4x16,f32) + C
  return __builtin_amdgcn_wmma_f32_16x16x4_f32(
      /*neg_a=*/false, a, /*neg_b=*/false, b,
      /*c_mod=*/(short)0, c, /*reuse_a=*/false, /*reuse_b=*/false);
}

// --------------------------------------------------------------------------
__global__ void zero_out_kernel(float* __restrict__ out, int n) {
  int i = blockIdx.x * blockDim.x + threadIdx.x;
  if (i < n) out[i] = 0.0f;
}

// --------------------------------------------------------------------------
// Generic column-strip projection GEMM:  Dst = act(A @ W + bias).
//   grid.x = 48  (16-column strips of the 768-wide output)
//   grid.y = m-groups; each block's 8 waves handle 8 consecutive 16-row tiles.
// The 768x16 weight strip (48 KB) is staged into LDS ONCE per block with the
// Tensor Data Mover, then shared by all 8 waves (8x L2-traffic reduction).
// mode 1: Dst = tanh(acc+bias) and also KH = tanh(Hsrc) elementwise.
__global__ __launch_bounds__(256) void proj_strip_kernel(
    const float* __restrict__ A, const float* __restrict__ W,
    const float* __restrict__ bias, float* __restrict__ Dst,
    const float* __restrict__ Hsrc, float* __restrict__ KH,
    int mValidRows, int mTiles, int mode) {
  __shared__ float Blds[D_MODEL * 16];      // 48 KB weight strip [k][n]

  const int tid  = threadIdx.x;
  const int wave = tid >> 5;
  const int lane = tid & 31;
  const int l16  = lane & 15;
  const int kh   = (lane >> 4) << 1;        // 0 or 2
  const int mhi  = (lane >> 4) * 8;

  const int nT   = blockIdx.x;              // column strip
  const int col0 = nT * 16;
  const int mT   = blockIdx.y * 8 + wave;   // this wave's row tile
  const int mTc  = mT < mTiles - 1 ? mT : mTiles - 1;
  const int row0 = mTc * 16;

  // ---- stage W[:, col0:col0+16] into LDS via the Tensor Data Mover ------
#if defined(__has_builtin) && __has_builtin(__builtin_amdgcn_tensor_load_to_lds)
  if (wave == 0) {
    const unsigned           lds_off = (unsigned)(size_t)(void*)Blds;
    const unsigned long long gaddr   = (unsigned long long)(size_t)(W + col0);
    // D# group 0: count=1 | lds_addr | global_addr[56:0] | type=2
    v4u g0;
    g0.x = 1u;
    g0.y = lds_off;
    g0.z = (unsigned)(gaddr & 0xffffffffu);
    g0.w = (unsigned)((gaddr >> 32) & 0x01ffffffu) | (2u << 30);
    // D# group 1: data_size=4B; tensor 768x768; tile 16(x) x 768(y); stride0=768
    v8i g1;
    g1[0] = (int)(2u << 16);                                   // data_size code 2
    g1[1] = (int)((D_MODEL & 0xffffu) << 16);                  // tensor_dim0 lo
    g1[2] = (int)(((unsigned)D_MODEL >> 16) |                  // tensor_dim0 hi
                  ((D_MODEL & 0xffffu) << 16));                // tensor_dim1 lo
    g1[3] = (int)(((unsigned)D_MODEL >> 16) | (16u << 16));    // dim1 hi | tile_dim0
    g1[4] = (int)(D_MODEL & 0xffffu);                          // tile_dim1 | tile_dim2=0
    g1[5] = (int)D_MODEL;                                      // tensor_dim0_stride lo
    g1[6] = (int)(((unsigned)(D_MODEL * D_MODEL) & 0xffffu) << 16); // stride1 lo16
    g1[7] = (int)((unsigned)(D_MODEL * D_MODEL) >> 16);        // stride1 hi
    v4i g2 = {0, 0, 0, 0};                  // dims 2-3 unused (2D tile)
    v4i g3 = {0, 0, 0, 0};
    v8i g4 = {0, 0, 0, 0, 0, 0, 0, 0};      // trailing group (unused)
    __builtin_amdgcn_tensor_load_to_lds(g0, g1, g2, g3, g4, 0);
    __builtin_amdgcn_s_wait_tensorcnt(0);
  }
#else
  for (int idx = tid; idx < D_MODEL * 16; idx += 256)
    Blds[idx] = W[(size_t)(idx >> 4) * D_MODEL + col0 + (idx & 15)];
#endif
  __syncthreads();

  // ---- 16x16 tile GEMM, K = 768 -----------------------------------------
  int rowA = row0 + l16;
  if (rowA > mValidRows - 1) rowA = mValidRows - 1;
  const float* Arow = A + (size_t)rowA * D_MODEL;
  v8f acc = {};
  for (int ks = 0; ks < D_MODEL / 4; ++ks) {
    const int k0 = ks * 4 + kh;
    v2f a = *(const v2f*)(Arow + k0);
    v2f b;
    b.x = Blds[k0 * 16 + l16];              // lanes 0-15: banks 0-15
    b.y = Blds[(k0 + 1) * 16 + l16];        // lanes 16-31: banks 32-47
    acc = wmma_f32(a, b, acc);
  }

  // ---- epilogue ----------------------------------------------------------
  const float bi = bias[col0 + l16];
  if (mT < mTiles) {
    for (int i = 0; i < 8; ++i) {
      const int r = row0 + mhi + i;
      if (r < mValidRows) {
        const size_t o = (size_t)r * D_MODEL + col0 + l16;
        if (mode == 1) {
          Dst[o] = tanhf(acc[i] + bi);
          KH[o]  = tanhf(Hsrc[o]);
        } else {
          Dst[o] = acc[i] + bi;
        }
      }
    }
  }
}

// --------------------------------------------------------------------------
// Fused attention: one block per (b, z, 16-row tile of (c,s)).
//   phase1: S[16][1024] = QS_tile @ KS_bz^T          (WMMA, K=64)
//   phase2: softmax over t (cross-wave via LDS)
//   phase3: ctx_partial = attn @ KH_bz               (WMMA, K split over waves)
//   phase4: reduce, mean over s, dot with Wql, atomicAdd into out[b][c]
__global__ __launch_bounds__(256) void attn_kernel(
    const float* __restrict__ QS, const float* __restrict__ KS,
    const float* __restrict__ KH, const float* __restrict__ WQL,
    float* __restrict__ out) {
  extern __shared__ float smem[];
  float* S       = smem;                    // 16 * SLD        (scores / attn)
  float* part    = S + 16 * SLD;            // 8 * 16 * 64     (per-wave ctx)
  float* ctx     = part + 8 * 16 * 64;      // 16 * 64
  float* red     = ctx + 16 * 64;           // 16 * 16
  float* rowstat = red + 16 * 16;           // 16

  const int tid  = threadIdx.x;
  const int wave = tid >> 5;
  const int lane = tid & 31;
  const int l16  = lane & 15;
  const int kh   = (lane >> 4) << 1;
  const int mhi  = (lane >> 4) * 8;

  const int rowTile = blockIdx.x % 13;      // 13 tiles cover 200 (c,s) rows
  const int bz      = blockIdx.x / 13;      // [0, 48)
  const int bb      = bz / NZ;
  const int zz      = bz % NZ;
  const int zbase   = zz * NH;
  const int row0    = rowTile * 16;

  // ---- phase 1: scores ---------------------------------------------------
  int rowA = row0 + l16; if (rowA > CS - 1) rowA = CS - 1;
  const float* Arow = QS + (size_t)rowA * D_MODEL + zbase;
  const int nb = wave * 128;                // this wave's t-range
  for (int nt = 0; nt < 8; ++nt) {
    v8f acc = {};
    const int tcol = nb + nt * 16 + l16;    // B-frag N index = t
    const float* Brow = KS + (size_t)(bb * NT + tcol) * D_MODEL + zbase;
    for (int ks = 0; ks < NH / 4; ++ks) {   // K = 64
      const int k0 = ks * 4 + kh;
      v2f a = *(const v2f*)(Arow + k0);
      v2f b = *(const v2f*)(Brow + k0);
      acc = wmma_f32(a, b, acc);
    }
    for (int i = 0; i < 8; ++i)
      S[(mhi + i) * SLD + tcol] = acc[i];
  }
  __syncthreads();

  // ---- phase 2: softmax over t=1024, per row -----------------------------
  {
    const int r = tid & 15;
    const int chunk = tid >> 4;             // 16 chunks of 64 columns
    float* Sr = S + r * SLD + chunk * 64;
    float m = -INFINITY;
    for (int j = 0; j < 64; ++j) m = fmaxf(m, Sr[j]);
    red[r * 16 + chunk] = m;
    __syncthreads();
    if (tid < 16) {
      float mm = -INFINITY;
      for (int j = 0; j < 16; ++j) mm = fmaxf(mm, red[tid * 16 + j]);
      rowstat[tid] = mm;
    }
    __syncthreads();
    const float rmax = rowstat[r];
    float ssum = 0.0f;
    for (int j = 0; j < 64; ++j) {
      const float e = __expf(Sr[j] - rmax);
      Sr[j] = e;
      ssum += e;
    }
    red[r * 16 + chunk] = ssum;
    __syncthreads();
    if (tid < 16) {
      float sm = 0.0f;
      for (int j = 0; j < 16; ++j) sm += red[tid * 16 + j];
      rowstat[tid] = 1.0f / sm;
    }
    __syncthreads();
    const float inv = rowstat[r];
    for (int j = 0; j < 64; ++j) Sr[j] *= inv;
  }
  __syncthreads();

  // ---- phase 3: ctx partials; each wave owns K range [wave*128, +128) ----
  const int kbase = wave * 128;
  for (int ht = 0; ht < 4; ++ht) {          // h in 4 tiles of 16
    v8f acc = {};
    const int hcol = ht * 16 + l16;
    for (int ks = 0; ks < 32; ++ks) {       // 128 / 4
      const int k0 = kbase + ks * 4 + kh;
      v2f a = *(const v2f*)(S + l16 * SLD + k0);          // attn from LDS
      v2f b;
      b.x = KH[(size_t)(bb * NT + k0)     * D_MODEL + zbase + hcol];
      b.y = KH[(size_t)(bb * NT + k0 + 1) * D_MODEL + zbase + hcol];
      acc = wmma_f32(a, b, acc);
    }
    for (int i = 0; i < 8; ++i)
      part[(wave * 16 + mhi + i) * 64 + hcol] = acc[i];
  }
  __syncthreads();

  // ---- phase 4a: reduce K-partials -> ctx[16][64] ------------------------
  for (int e = tid; e < 16 * 64; e += 256) {
    float v = 0.0f;
    for (int w = 0; w < 8; ++w) v += part[w * 1024 + e];
    ctx[e] = v;
  }
  __syncthreads();

  // ---- phase 4b: mean over s, dot with Wql, accumulate over z ------------
  if (wave < 4) {                           // 4 c-values in this row tile
    const int cloc  = wave;
    const int cglob = rowTile * 4 + cloc;
    if (cglob < NC) {
      const float* wq = WQL + (size_t)cglob * D_MODEL + zbase;
      float acc = 0.0f;
      for (int j = 0; j < 8; ++j) {
        const int idx  = lane + 32 * j;     // 0..255 over (s, h)
        const int srow = idx >> 6;
        const int hh   = idx & 63;
        acc += ctx[(cloc * 4 + srow) * 64 + hh] * wq[hh];
      }
      for (int off = 16; off > 0; off >>= 1)
        acc += __shfl_xor(acc, off, 32);    // wave32 reduction
      if (lane == 0)
        atomicAdd(out + bb * NC + cglob, 0.25f * acc);  // 0.25 = mean over s
    }
  }
}

// --------------------------------------------------------------------------
extern "C" void kernel_launch(void* const* d_in, const int* in_sizes, int n_in,
                              void* d_out, int out_size, void* d_ws,
                              size_t ws_size, hipStream_t stream) {
  (void)in_sizes; (void)n_in; (void)ws_size;
  const float* Q   = (const float*)d_in[0];
  const float* H   = (const float*)d_in[1];
  const float* ql  = (const float*)d_in[2];
  const float* WQw = (const float*)d_in[3];
  const float* WQb = (const float*)d_in[4];
  const float* WKw = (const float*)d_in[5];
  const float* WKb = (const float*)d_in[6];
  const float* WVw = (const float*)d_in[7];
  const float* WVb = (const float*)d_in[8];
  float* out = (float*)d_out;

  float* ws  = (float*)d_ws;
  float* KS  = ws;                                    // 4096 * 768
  float* KH  = KS + (size_t)NBATCH * NT * D_MODEL;    // 4096 * 768
  float* QS  = KH + (size_t)NBATCH * NT * D_MODEL;    // 200 * 768
  float* WQL = QS + (size_t)CS * D_MODEL;             // 50 * 768

  zero_out_kernel<<<1, 256, 0, stream>>>(out, out_size);

  // KS = tanh(H @ WK + bk), KH = tanh(H):  M=4096 -> 256 m-tiles, 32 m-groups
  proj_strip_kernel<<<dim3(48, 32), 256, 0, stream>>>(
      H, WKw, WKb, KS, H, KH, NBATCH * NT, 256, /*mode=*/1);
  // QS = Q @ WQ + bq:  M=200 -> 13 m-tiles, 2 m-groups
  proj_strip_kernel<<<dim3(48, 2), 256, 0, stream>>>(
      Q, WQw, WQb, QS, nullptr, nullptr, CS, 13, /*mode=*/0);
  // WQL = ql @ WV + bv:  M=50 -> 4 m-tiles, 1 m-group
  proj_strip_kernel<<<dim3(48, 1), 256, 0, stream>>>(
      ql, WVw, WVb, WQL, nullptr, nullptr, NC, 4, /*mode=*/0);

  const size_t smem =
      (size_t)(16 * SLD + 8 * 16 * 64 + 16 * 64 + 16 * 16 + 16) * sizeof(float);
  attn_kernel<<<624, 256, smem, stream>>>(QS, KS, KH, WQL, out);
}